// LinearGaussianSSM_37615323578933
// MI455X (gfx1250) — compile-verified
//
#include <hip/hip_runtime.h>
#include <hip/hip_bf16.h>

// ---------------------------------------------------------------------------
// Persistent-kernel Kalman filter for MI455X (gfx1250, wave32).
//   STATE=256, OBS=128, T=2048.
// - All large GEMMs: V_WMMA_F32_16X16X4_F32 (full-f32 matrix op; the 2048-step
//   covariance recursion is not viable in bf16/f16).
// - Cholesky + L^-1 run entirely in LDS (128 KB of the WGP's 320 KB), so the
//   128 serial factorization steps bounce off DS latency instead of L2.
// - 16 workgroups x 256 threads persistent; phases separated by a software
//   grid barrier (device-scope atomics in d_ws, counters zeroed per launch).
// ---------------------------------------------------------------------------

#define KF_STATE 256
#define KF_OBS   128
#define KF_T     2048
#define KF_NWG   16
#define KF_BLOCK 256
#define KF_WAVES (KF_NWG * KF_BLOCK / 32)   // 128 waves
#define KF_LOG2PI 1.8378770664093453f

typedef float v2f __attribute__((ext_vector_type(2)));
typedef float v8f __attribute__((ext_vector_type(8)));

// ---- scratch layout (floats) ----------------------------------------------
#define OFF_P     64        // 256x256 predicted covariance
#define OFF_HP    65600     // 128x256 H*P
#define OFF_S     98368     // 128x128 innovation cov (input to WG0 Cholesky)
#define OFF_LINV  114752    // 128x128 L^-1 (published by WG0 for the GEMMs)
#define OFF_W1    131136    // 128x256 L^-1 * HP
#define OFF_KT    163904    // 128x256 Kt = S^-1 * HP
#define OFF_PNEW  196672    // 256x256 unsymmetrized filtered covariance
#define OFF_T1    262208    // 256x256 F * Psym
#define OFF_MP    327744    // 256 predicted mean
#define OFF_E     328000    // 128 innovation
#define OFF_U     328128    // 128 whitened innovation u = L^-1 e

// ---------------------------------------------------------------------------
// Grid barrier: release/acquire over device-scope atomics. cnt is zeroed by
// hipMemsetAsync before every launch; gen may hold any value (sense captured
// per barrier). The ACQ_REL fetch_add's release half keeps the preceding
// relaxed gen-load from sinking past the arrival.
// ---------------------------------------------------------------------------
__device__ __forceinline__ void grid_barrier(unsigned* cnt, unsigned* gen) {
  __threadfence();
  __syncthreads();
  if (threadIdx.x == 0) {
    unsigned g = __hip_atomic_load(gen, __ATOMIC_RELAXED, __HIP_MEMORY_SCOPE_AGENT);
    unsigned a = __hip_atomic_fetch_add(cnt, 1u, __ATOMIC_ACQ_REL, __HIP_MEMORY_SCOPE_AGENT);
    if (a == (unsigned)(gridDim.x - 1)) {
      __hip_atomic_store(cnt, 0u, __ATOMIC_RELAXED, __HIP_MEMORY_SCOPE_AGENT);
      __hip_atomic_fetch_add(gen, 1u, __ATOMIC_RELEASE, __HIP_MEMORY_SCOPE_AGENT);
    } else {
      while (__hip_atomic_load(gen, __ATOMIC_ACQUIRE, __HIP_MEMORY_SCOPE_AGENT) == g) {
        __builtin_amdgcn_s_sleep(1);
      }
    }
  }
  __syncthreads();
  __threadfence();
}

// ---------------------------------------------------------------------------
// Generic f32 WMMA GEMM:  D = (negA ? -1 : 1) * op(A) @ op(B) + (C ? C : 0)
// 16x16 tiles striped over all waves in the grid; K stepped by 4 using
// V_WMMA_F32_16X16X4_F32. ISA 7.12.2 fragment layouts:
//   A (16x4):  lane l holds A[l&15][k0 + 2*(l>>4) + j]   in a[j]
//   B (4x16):  lane l holds B[k0 + 2*(l>>4) + j][l&15]   in b[j]
//   C/D:       acc[r] <-> row r + 8*(l>>4), col l&15
// Tile/K loop bounds are wave-uniform so EXEC stays all-ones for WMMA.
// ---------------------------------------------------------------------------
__device__ __forceinline__ void gemm_f32_wmma(
    const float* __restrict__ A, int lda, bool tA, bool negA,
    const float* __restrict__ B, int ldb, bool tB,
    const float* __restrict__ C, int ldc,
    float* __restrict__ D, int ldd,
    int M, int N, int K, int wid, int lane) {
  const int half = lane >> 4;   // 0/1
  const int l16  = lane & 15;
  const int ntN  = N >> 4;
  const int ntiles = (M >> 4) * ntN;
  for (int tile = wid; tile < ntiles; tile += KF_WAVES) {
    const int tm = (tile / ntN) << 4;
    const int tn = (tile % ntN) << 4;
    const int ra = tm + l16;   // A row handled by this lane
    const int cb = tn + l16;   // B col handled by this lane
    v8f acc;
    if (C) {
#pragma unroll
      for (int r = 0; r < 8; ++r)
        acc[r] = C[(tm + r + (half << 3)) * ldc + tn + l16];
    } else {
#pragma unroll
      for (int r = 0; r < 8; ++r) acc[r] = 0.0f;
    }
    for (int k0 = 0; k0 < K; k0 += 4) {
      const int kb = k0 + (half << 1);
      v2f a, b;
#pragma unroll
      for (int j = 0; j < 2; ++j) {
        const int k = kb + j;
        float av = tA ? A[k * lda + ra] : A[ra * lda + k];
        a[j] = negA ? -av : av;
        b[j] = tB ? B[cb * ldb + k] : B[k * ldb + cb];
      }
      acc = __builtin_amdgcn_wmma_f32_16x16x4_f32(
          false, a, false, b, (short)0, acc, false, false);
    }
#pragma unroll
    for (int r = 0; r < 8; ++r)
      D[(tm + r + (half << 3)) * ldd + tn + l16] = acc[r];
  }
}

// ---------------------------------------------------------------------------
__global__ void __launch_bounds__(KF_BLOCK)
kf_persistent_kernel(const float* __restrict__ emissions,
                     const float* __restrict__ m0,
                     const float* __restrict__ P0,
                     const float* __restrict__ F,
                     const float* __restrict__ Q,
                     const float* __restrict__ H,
                     const float* __restrict__ R,
                     float* __restrict__ out,
                     void* __restrict__ wsv) {
  float*    ws   = (float*)wsv;
  unsigned* cnt  = (unsigned*)wsv;       // [0]
  unsigned* gen  = (unsigned*)wsv + 1;   // [1]

  float* P    = ws + OFF_P;
  float* HP   = ws + OFF_HP;
  float* Smat = ws + OFF_S;
  float* Linv = ws + OFF_LINV;
  float* W1   = ws + OFF_W1;
  float* Kt   = ws + OFF_KT;
  float* Pnew = ws + OFF_PNEW;
  float* T1   = ws + OFF_T1;
  float* mp   = ws + OFF_MP;
  float* evec = ws + OFF_E;
  float* uvec = ws + OFF_U;

  float* means = out + 1;
  float* covs  = out + 1 + (size_t)KF_T * KF_STATE;

  const int tid  = threadIdx.x;
  const int bid  = blockIdx.x;
  const int gtid = bid * KF_BLOCK + tid;
  const int lane = tid & 31;
  const int wid  = gtid >> 5;
  const int gsz  = KF_NWG * KF_BLOCK;

  // 129 KB of the WGP's 320 KB LDS: the entire Cholesky working set.
  __shared__ float Ssh[KF_OBS * KF_OBS];   // 64 KB: S -> L (lower)
  __shared__ float Lsh[KF_OBS * KF_OBS];   // 64 KB: L^-1
  __shared__ float sred[KF_BLOCK];

  float ll = 0.0f;   // meaningful on (block 0, thread 0) only

  // ---- init state from inputs ----
  for (int idx = gtid; idx < KF_STATE * KF_STATE; idx += gsz) P[idx] = P0[idx];
  for (int idx = gtid; idx < KF_STATE; idx += gsz)            mp[idx] = m0[idx];
  grid_barrier(cnt, gen);

  for (int t = 0; t < KF_T; ++t) {
    const float* y = emissions + (size_t)t * KF_OBS;

    // ---- Phase A: HP = H @ P ; WG1: e = y - H @ mp ----
    gemm_f32_wmma(H, KF_STATE, false, false, P, KF_STATE, false,
                  (const float*)nullptr, 0, HP, KF_STATE,
                  KF_OBS, KF_STATE, KF_STATE, wid, lane);
    if (bid == 1) {
      for (int i = tid; i < KF_OBS; i += KF_BLOCK) {
        float acc = y[i];
        for (int k = 0; k < KF_STATE; ++k) acc -= H[i * KF_STATE + k] * mp[k];
        evec[i] = acc;
      }
      if (t + 1 < KF_T && tid < 8) {     // warm L2 for next step's row
        __builtin_prefetch(emissions + (size_t)(t + 1) * KF_OBS + tid * 16, 0, 1);
      }
    }
    grid_barrier(cnt, gen);

    // ---- Phase B: S = HP @ H^T + R ----
    gemm_f32_wmma(HP, KF_STATE, false, false, H, KF_STATE, true,
                  R, KF_OBS, Smat, KF_OBS,
                  KF_OBS, KF_OBS, KF_STATE, wid, lane);
    grid_barrier(cnt, gen);

    // ---- Phase C (WG0, in LDS): Cholesky(S)->L, Linv=L^-1, u=Linv@e, ll ----
    if (bid == 0) {
      for (int idx = tid; idx < KF_OBS * KF_OBS; idx += KF_BLOCK)
        Ssh[idx] = Smat[idx];
      __syncthreads();
      // in-place right-looking Cholesky (lower triangle of Ssh)
      for (int j = 0; j < KF_OBS; ++j) {
        if (tid == 0) Ssh[j * KF_OBS + j] = sqrtf(Ssh[j * KF_OBS + j]);
        __syncthreads();
        const float d = Ssh[j * KF_OBS + j];
        for (int i = j + 1 + tid; i < KF_OBS; i += KF_BLOCK)
          Ssh[i * KF_OBS + j] /= d;
        __syncthreads();
        const int rem = KF_OBS - 1 - j;
        for (int idx = tid; idx < rem * rem; idx += KF_BLOCK) {
          const int i = j + 1 + idx / rem;
          const int k = j + 1 + idx % rem;
          Ssh[i * KF_OBS + k] -= Ssh[i * KF_OBS + j] * Ssh[k * KF_OBS + j];
        }
        __syncthreads();
      }
      // Lsh = I, then right-looking forward solve  L * Lsh = I
      for (int idx = tid; idx < KF_OBS * KF_OBS; idx += KF_BLOCK)
        Lsh[idx] = ((idx >> 7) == (idx & 127)) ? 1.0f : 0.0f;
      __syncthreads();
      for (int j = 0; j < KF_OBS; ++j) {
        const float d = Ssh[j * KF_OBS + j];
        for (int c = tid; c <= j; c += KF_BLOCK) Lsh[j * KF_OBS + c] /= d;
        __syncthreads();
        const int rem = KF_OBS - 1 - j;
        for (int idx = tid; idx < rem * KF_OBS; idx += KF_BLOCK) {
          const int i = j + 1 + (idx >> 7);
          const int c = idx & 127;
          Lsh[i * KF_OBS + c] -= Ssh[i * KF_OBS + j] * Lsh[j * KF_OBS + c];
        }
        __syncthreads();
      }
      // publish L^-1 for the cross-workgroup GEMMs (one coalesced pass)
      for (int idx = tid; idx < KF_OBS * KF_OBS; idx += KF_BLOCK)
        Linv[idx] = Lsh[idx];
      // u = L^-1 @ e (lower-triangular), q = u.u, logdet = 2*sum(log diag L)
      float uacc = 0.0f;
      if (tid < KF_OBS) {
        for (int k = 0; k <= tid; ++k)
          uacc += Lsh[tid * KF_OBS + k] * evec[k];
        uvec[tid] = uacc;
      }
      sred[tid] = (tid < KF_OBS) ? uacc * uacc : 0.0f;
      __syncthreads();
      for (int s = KF_BLOCK / 2; s >= 1; s >>= 1) {
        if (tid < s) sred[tid] += sred[tid + s];
        __syncthreads();
      }
      const float qv = sred[0];
      __syncthreads();
      sred[tid] = (tid < KF_OBS) ? logf(Ssh[tid * KF_OBS + tid]) : 0.0f;
      __syncthreads();
      for (int s = KF_BLOCK / 2; s >= 1; s >>= 1) {
        if (tid < s) sred[tid] += sred[tid + s];
        __syncthreads();
      }
      if (tid == 0)
        ll += -0.5f * ((float)KF_OBS * KF_LOG2PI + 2.0f * sred[0] + qv);
    }
    grid_barrier(cnt, gen);

    // ---- Phase D: W1 = Linv @ HP ----
    gemm_f32_wmma(Linv, KF_OBS, false, false, HP, KF_STATE, false,
                  (const float*)nullptr, 0, W1, KF_STATE,
                  KF_OBS, KF_STATE, KF_OBS, wid, lane);
    grid_barrier(cnt, gen);

    // ---- Phase E: Kt = Linv^T @ W1 ; WG1: m, means[t], mp = F@m ----
    gemm_f32_wmma(Linv, KF_OBS, true, false, W1, KF_STATE, false,
                  (const float*)nullptr, 0, Kt, KF_STATE,
                  KF_OBS, KF_STATE, KF_OBS, wid, lane);
    if (bid == 1) {
      {
        float acc = mp[tid];                     // m = mp + W1^T u
        for (int k = 0; k < KF_OBS; ++k) acc += W1[k * KF_STATE + tid] * uvec[k];
        sred[tid] = acc;
        means[(size_t)t * KF_STATE + tid] = acc;
      }
      __syncthreads();
      {
        float mn = 0.0f;                          // mp_next = F @ m
        for (int k = 0; k < KF_STATE; ++k) mn += F[tid * KF_STATE + k] * sred[k];
        mp[tid] = mn;
      }
    }
    grid_barrier(cnt, gen);

    // ---- Phase F: Pnew = P - HP^T @ Kt  (A = -HP^T, C-init = P) ----
    gemm_f32_wmma(HP, KF_STATE, true, true, Kt, KF_STATE, false,
                  P, KF_STATE, Pnew, KF_STATE,
                  KF_STATE, KF_STATE, KF_OBS, wid, lane);
    grid_barrier(cnt, gen);

    // ---- Phase G: symmetrize -> covs[t] (also serves as Psym buffer) ----
    float* ct = covs + (size_t)t * KF_STATE * KF_STATE;
    for (int idx = gtid; idx < KF_STATE * KF_STATE; idx += gsz) {
      const int i = idx >> 8, j = idx & 255;
      ct[idx] = 0.5f * (Pnew[idx] + Pnew[(j << 8) + i]);
    }
    grid_barrier(cnt, gen);

    // ---- Phase H: T1 = F @ Psym ----
    gemm_f32_wmma(F, KF_STATE, false, false, ct, KF_STATE, false,
                  (const float*)nullptr, 0, T1, KF_STATE,
                  KF_STATE, KF_STATE, KF_STATE, wid, lane);
    grid_barrier(cnt, gen);

    // ---- Phase I: P = T1 @ F^T + Q ----
    gemm_f32_wmma(T1, KF_STATE, false, false, F, KF_STATE, true,
                  Q, KF_STATE, P, KF_STATE,
                  KF_STATE, KF_STATE, KF_STATE, wid, lane);
    grid_barrier(cnt, gen);
  }

  if (bid == 0 && tid == 0) out[0] = ll;
}

// ---------------------------------------------------------------------------
extern "C" void kernel_launch(void* const* d_in, const int* in_sizes, int n_in,
                              void* d_out, int out_size, void* d_ws, size_t ws_size,
                              hipStream_t stream) {
  const float* emissions = (const float*)d_in[0];
  const float* m0        = (const float*)d_in[1];
  const float* P0        = (const float*)d_in[2];
  const float* F         = (const float*)d_in[3];
  const float* Q         = (const float*)d_in[4];
  const float* H         = (const float*)d_in[5];
  const float* R         = (const float*)d_in[6];

  // zero the barrier counters (capture-legal stream op)
  hipMemsetAsync(d_ws, 0, 256, stream);

  kf_persistent_kernel<<<KF_NWG, KF_BLOCK, 0, stream>>>(
      emissions, m0, P0, F, Q, H, R, (float*)d_out, d_ws);
}